// VarianceimpFeatureLoss_23880018166010
// MI455X (gfx1250) — compile-verified
//
#include <hip/hip_runtime.h>
#include <hip/hip_bf16.h>
#include <stdint.h>

// ---------------------------------------------------------------------------
// VarianceimpFeatureLoss for [8,3,512,512] fp32 preds/target -> scalar fp32.
//
// MI455X strategy (compile-only reasoning):
//  * Roofline: 50 MB reads -> ~2.2us at 23.3 TB/s; selection networks dominate
//    (VALU-bound). So: (1) stage tiles once in LDS via the Tensor Data Mover
//    (one tensor_load_to_lds per tensor per workgroup, TENSORcnt-tracked),
//    (2) minimize min/max ops via shared column sorts + bitonic partial
//    merges (~72 ops/window instead of ~224 for naive insertion).
//  * wave32: 256-thread blocks = 8 waves; ~35KB LDS/WG of 320KB per WGP.
//  * Deterministic two-stage reduction through d_ws (no float atomics).
// ---------------------------------------------------------------------------

#define TS     64          // output tile (window origins / variance region)
#define HALO   67          // TS + K - 1 input rows/cols needed
#define SW     67          // LDS row stride in floats (matches TDM row packing)
#define NBLK   1536        // 24 planes * 8*8 tiles
#define PLANE  (512 * 512)

typedef __attribute__((ext_vector_type(4))) unsigned int u32x4;
typedef __attribute__((ext_vector_type(8))) unsigned int u32x8;

// Generic-space LDS pointer -> LDS byte address (low 32 bits of the generic address).
__device__ __forceinline__ uint32_t lds_addr32(const void* p) {
    return (uint32_t)(uintptr_t)p;
}

// Issue one TDM 2D tile load: global (row stride 512 floats) -> LDS, tile
// HALO x HALO elements. tensor_dim0/1 = remaining extent from the tile origin
// so out-of-tensor reads at right/bottom edges return zero (those halo values
// are only touched by masked-off windows). D# layout per CDNA5 ISA sec. 8.3
// (group0) / 8.4 (group1); groups 2/3 NULL (2D tensor).
__device__ __forceinline__ void tdm_load_2d(uint32_t lds, const float* g,
                                            uint32_t tdim0, uint32_t tdim1) {
    uint64_t ga = (uint64_t)(uintptr_t)g;
    u32x4 g0;
    g0[0] = 1u;                                                // count=1 (valid D#)
    g0[1] = lds;                                               // lds_addr (bytes)
    g0[2] = (uint32_t)ga;                                      // global_addr[31:0]
    g0[3] = ((uint32_t)(ga >> 32) & 0x01FFFFFFu) | (2u << 30); // addr[56:32] | type=2
    u32x8 g1;
    g1[0] = 2u << 16;                                          // data_size=2 (4B), mask=0
    g1[1] = (tdim0 & 0xFFFFu) << 16;                           // tensor_dim0[15:0]
    g1[2] = ((tdim0 >> 16) & 0xFFFFu) | ((tdim1 & 0xFFFFu) << 16); // dim0 hi | dim1 lo
    g1[3] = ((tdim1 >> 16) & 0xFFFFu) | ((uint32_t)HALO << 16);    // dim1 hi | tile_dim0
    g1[4] = (uint32_t)HALO;                                    // tile_dim1 | tile_dim2=0
    g1[5] = 512u;                                              // tensor_dim0_stride lo32
    g1[6] = 0u;                                                // stride0 hi | stride1 lo
    g1[7] = 0u;                                                // stride1 hi
    asm volatile("tensor_load_to_lds %0, %1" : : "s"(g0), "s"(g1) : "memory");
}

__device__ __forceinline__ void wait_tdm0() {
#if __has_builtin(__builtin_amdgcn_s_wait_tensorcnt)
    __builtin_amdgcn_s_wait_tensorcnt(0);
#else
    asm volatile("s_wait_tensorcnt 0" ::: "memory");
#endif
}

// ---- selection-network primitives -----------------------------------------

// compare-exchange, descending (a keeps the larger value)
__device__ __forceinline__ void ced(float& a, float& b) {
    float t = fminf(a, b); a = fmaxf(a, b); b = t;
}
// compare-exchange, ascending (a keeps the smaller value)
__device__ __forceinline__ void cea(float& a, float& b) {
    float t = fmaxf(a, b); a = fminf(a, b); b = t;
}
// 5-CE sorting network, descending
__device__ __forceinline__ void sort4d(float& a, float& b, float& c, float& d) {
    ced(a, b); ced(c, d); ced(a, c); ced(b, d); ced(b, c);
}
// A,B desc-sorted quads. After this, A = desc-sorted top-4 of A∪B.
// {max(a_i, b_{3-i})} is the top-4 multiset (bitonic); 4-CE half-clean sorts it.
__device__ __forceinline__ void merge_top4(float& a0, float& a1, float& a2, float& a3,
                                           float b0, float b1, float b2, float b3) {
    a0 = fmaxf(a0, b3); a1 = fmaxf(a1, b2); a2 = fmaxf(a2, b1); a3 = fmaxf(a3, b0);
    ced(a0, a2); ced(a1, a3); ced(a0, a1); ced(a2, a3);
}
// A,B asc-sorted quads. After this, A = asc-sorted bottom-4 of A∪B.
__device__ __forceinline__ void merge_bot4(float& a0, float& a1, float& a2, float& a3,
                                           float b0, float b1, float b2, float b3) {
    a0 = fminf(a0, b3); a1 = fminf(a1, b2); a2 = fminf(a2, b1); a3 = fminf(a3, b0);
    cea(a0, a2); cea(a1, a3); cea(a0, a1); cea(a2, a3);
}

// Process 4 horizontally consecutive 4x4 windows starting at `base`
// (row stride SW). Shares the 7 sorted vertical columns across windows.
// tsum[w]/bsum[w] = sum of top-4 / bottom-4 of window w.
__device__ __forceinline__ void group4(const float* __restrict__ base,
                                       float tsum[4], float bsum[4]) {
    float c[7][4];                       // desc-sorted columns, kept in VGPRs
#pragma unroll
    for (int j = 0; j < 7; ++j) {
        float v0 = base[j];
        float v1 = base[j + SW];
        float v2 = base[j + 2 * SW];
        float v3 = base[j + 3 * SW];
        sort4d(v0, v1, v2, v3);
        c[j][0] = v0; c[j][1] = v1; c[j][2] = v2; c[j][3] = v3;
    }
#pragma unroll
    for (int w = 0; w < 4; ++w) {
        // top-4 path (desc views)
        float t0 = c[w][0], t1 = c[w][1], t2 = c[w][2], t3 = c[w][3];
        merge_top4(t0, t1, t2, t3, c[w + 1][0], c[w + 1][1], c[w + 1][2], c[w + 1][3]);
        float s0 = c[w + 2][0], s1 = c[w + 2][1], s2 = c[w + 2][2], s3 = c[w + 2][3];
        merge_top4(s0, s1, s2, s3, c[w + 3][0], c[w + 3][1], c[w + 3][2], c[w + 3][3]);
        // final level: only the sum is needed -> no re-sort
        t0 = fmaxf(t0, s3); t1 = fmaxf(t1, s2); t2 = fmaxf(t2, s1); t3 = fmaxf(t3, s0);
        tsum[w] = (t0 + t1) + (t2 + t3);
        // bottom-4 path (asc views = reversed desc columns)
        float u0 = c[w][3], u1 = c[w][2], u2 = c[w][1], u3 = c[w][0];
        merge_bot4(u0, u1, u2, u3, c[w + 1][3], c[w + 1][2], c[w + 1][1], c[w + 1][0]);
        float r0 = c[w + 2][3], r1 = c[w + 2][2], r2 = c[w + 2][1], r3 = c[w + 2][0];
        merge_bot4(r0, r1, r2, r3, c[w + 3][3], c[w + 3][2], c[w + 3][1], c[w + 3][0]);
        u0 = fminf(u0, r3); u1 = fminf(u1, r2); u2 = fminf(u2, r1); u3 = fminf(u3, r0);
        bsum[w] = (u0 + u1) + (u2 + u3);
    }
}

__global__ void __launch_bounds__(256)
vifl_tile_kernel(const float* __restrict__ preds,
                 const float* __restrict__ target,
                 float* __restrict__ part) {
    __shared__ float sp[HALO * SW];   // preds tile  (4489 floats)
    __shared__ float st[HALO * SW];   // target tile

    const int tid   = threadIdx.x;
    const int plane = blockIdx.x >> 6;        // 24 planes
    const int tile  = blockIdx.x & 63;        // 8x8 tiles per plane
    const int y0    = (tile >> 3) << 6;
    const int x0    = (tile & 7)  << 6;
    const size_t pb = (size_t)plane * PLANE + (size_t)y0 * 512 + x0;

    // --- Stage both tiles into LDS with the Tensor Data Mover (wave 0 only;
    //     TDM is a wave-level op, EXEC-independent). ------------------------
    if (tid < 32) {
        tdm_load_2d(lds_addr32(sp), preds  + pb, 512u - x0, 512u - y0);
        tdm_load_2d(lds_addr32(st), target + pb, 512u - x0, 512u - y0);
        wait_tdm0();
    }
    __syncthreads();

    // --- Sliding 4x4 windows: 64x64 origins per tile, processed as 1024
    //     groups of 4 consecutive-in-x windows; 4 groups per thread. -------
    float accMax = 0.f, accMin = 0.f;
    const int oyMax = 508 - y0;   // window origin must satisfy y0+oy <= 508
    const int oxMax = 508 - x0;
    for (int k = 0; k < 4; ++k) {
        const int g   = tid + (k << 8);   // 0..1023
        const int oy  = g >> 4;           // 0..63
        const int ox4 = (g & 15) << 2;    // 0,4,...,60  (reads stay <= col 66)
        float tp[4], bp[4], tt[4], bt[4];
        group4(&sp[oy * SW + ox4], tp, bp);
        group4(&st[oy * SW + ox4], tt, bt);
        const bool rowOk = (oy <= oyMax);
#pragma unroll
        for (int w = 0; w < 4; ++w) {
            if (rowOk && (ox4 + w) <= oxMax) {
                accMax += fabsf(tp[w] - tt[w]);   // 4 * |max4_p - max4_t|
                accMin += fabsf(bp[w] - bt[w]);   // 4 * |min4_p - min4_t|
            }
        }
    }

    // --- Non-overlapping 4x4 variance blocks: 16x16 per tile, 1/thread. ---
    const int by = (tid >> 4) << 2;
    const int bx = (tid & 15) << 2;
    float s0 = 0.f, q0 = 0.f, s1 = 0.f, q1 = 0.f;
#pragma unroll
    for (int r = 0; r < 4; ++r)
#pragma unroll
        for (int c = 0; c < 4; ++c) {
            const float vp = sp[(by + r) * SW + bx + c];
            const float vt = st[(by + r) * SW + bx + c];
            s0 += vp; q0 += vp * vp;
            s1 += vt; q1 += vt * vt;
        }
    const float mp = s0 * 0.0625f, mt = s1 * 0.0625f;
    const float accVar = fabsf((q0 * 0.0625f - mp * mp) -
                               (q1 * 0.0625f - mt * mt));

    // --- Block reduction (reuse sp) and deterministic partial store. ------
    __syncthreads();
    sp[tid]       = accVar;
    sp[256 + tid] = accMax;
    sp[512 + tid] = accMin;
    __syncthreads();
    for (int s = 128; s > 0; s >>= 1) {
        if (tid < s) {
            sp[tid]       += sp[tid + s];
            sp[256 + tid] += sp[256 + tid + s];
            sp[512 + tid] += sp[512 + tid + s];
        }
        __syncthreads();
    }
    if (tid == 0) {
        part[3 * blockIdx.x + 0] = sp[0];
        part[3 * blockIdx.x + 1] = sp[256];
        part[3 * blockIdx.x + 2] = sp[512];
    }
}

__global__ void __launch_bounds__(256)
vifl_finalize_kernel(const float* __restrict__ part, float* __restrict__ out) {
    __shared__ float r0[256], r1[256], r2[256];
    const int tid = threadIdx.x;
    float v = 0.f, mx = 0.f, mn = 0.f;
    for (int i = tid; i < NBLK; i += 256) {
        v  += part[3 * i + 0];
        mx += part[3 * i + 1];
        mn += part[3 * i + 2];
    }
    r0[tid] = v; r1[tid] = mx; r2[tid] = mn;
    __syncthreads();
    for (int s = 128; s > 0; s >>= 1) {
        if (tid < s) {
            r0[tid] += r0[tid + s];
            r1[tid] += r1[tid + s];
            r2[tid] += r2[tid + s];
        }
        __syncthreads();
    }
    if (tid == 0) {
        const float invVar = 1.0f / 393216.0f;     // 24 * 128 * 128
        const float invWin = 0.25f / 6217944.0f;   // /4 (sum->mean of 4) / (24*509*509)
        out[0] = r0[0] * invVar + r1[0] * invWin + r2[0] * invWin;
    }
}

extern "C" void kernel_launch(void* const* d_in, const int* in_sizes, int n_in,
                              void* d_out, int out_size, void* d_ws, size_t ws_size,
                              hipStream_t stream) {
    const float* preds  = (const float*)d_in[0];
    const float* target = (const float*)d_in[1];
    float* part = (float*)d_ws;   // needs NBLK*3*4 = 18432 bytes

    vifl_tile_kernel<<<dim3(NBLK), dim3(256), 0, stream>>>(preds, target, part);
    vifl_finalize_kernel<<<dim3(1), dim3(256), 0, stream>>>(part, (float*)d_out);
}